// SplatterBlender_19146964206318
// MI455X (gfx1250) — compile-verified
//
#include <hip/hip_runtime.h>
#include <stdint.h>

// ---------------------------------------------------------------------------
// SplatterBlender for gfx1250 (MI455X).
// Per-block: stage raw halo tiles into LDS via Tensor Data Mover (interior
// blocks), precompute separable Gaussian factors + masked depth per point in
// LDS, then each thread composites its pixel from the 9-neighborhood.
// ---------------------------------------------------------------------------

#define TILE_X 16
#define TILE_Y 8
#define HX (TILE_X + 2)          // 18
#define HY (TILE_Y + 2)          // 10
#define NHALO (HX * HY)          // 180
#define PSTRIDE 37               // 4*(3 wx + 3 wy + rgb) = 36, +1 pad (odd -> conflict-free LDS reads)

static constexpr int N_ = 2, H_ = 512, W_ = 512, K_ = 4;

// LDS layout (float offsets)
#define DEP_FLOATS  (NHALO * 4)                  // depths, float4-aligned per pixel
#define PRE_OFF     DEP_FLOATS
#define PRE_FLOATS  (NHALO * PSTRIDE)            // 6660
#define RAWC_OFF    (PRE_OFF + PRE_FLOATS)       // 7380
#define RAWC_FLOATS (NHALO * 12)                 // colors: 12 f32 / pixel
#define RAWP_OFF    (RAWC_OFF + RAWC_FLOATS)
#define RAWP_FLOATS (NHALO * 12)                 // pixel_coords_screen: 12 f32 / pixel
#define RAWM_OFF    (RAWP_OFF + RAWP_FLOATS)
#define RAWM_INTS   (NHALO * 4)                  // mask: 4 i32 / pixel
#define SMEM_FLOATS (RAWM_OFF + RAWM_INTS)       // 12420 floats
#define SMEM_BYTES  (SMEM_FLOATS * 4)            // 49680 B (< 64KB; 320KB/WGP available)

// ---- CDNA5 Tensor Data Mover availability ---------------------------------
#if defined(__has_builtin)
#  if __has_builtin(__builtin_amdgcn_tensor_load_to_lds) && __has_builtin(__builtin_amdgcn_s_wait_tensorcnt)
#    define USE_TDM 1
#  endif
#endif
#ifndef USE_TDM
#  define USE_TDM 0
#endif

#if USE_TDM
typedef unsigned int u32x4 __attribute__((ext_vector_type(4)));
typedef int i32x8 __attribute__((ext_vector_type(8)));
typedef int i32x4 __attribute__((ext_vector_type(4)));

// Issue one 2-D TDM tile load: global (strided rows) -> LDS (packed rows).
// D# packing per CDNA5 ISA ch.8 (group0: 128b, group1: 256b). data_size = 4B.
__device__ __forceinline__ void tdm_load_2d(unsigned lds_byte_off, const void* gptr,
                                            unsigned tile_elems, unsigned tile_rows,
                                            unsigned long long row_stride_elems)
{
  const unsigned long long ga = (unsigned long long)gptr;
  u32x4 g0;
  g0[0] = 1u;                                                 // count=1 (valid user D#), gather off
  g0[1] = lds_byte_off;                                       // lds_addr (bytes)
  g0[2] = (unsigned)(ga & 0xFFFFFFFFu);                       // global_addr[31:0]
  g0[3] = (unsigned)((ga >> 32) & 0x01FFFFFFu) | (2u << 30);  // global_addr[56:32], type=2 (image)

  const unsigned td0 = 0x40000000u;   // huge tensor dims: interior tiles never OOB
  const unsigned td1 = 0x40000000u;
  i32x8 g1;
  g1[0] = (int)(2u << 16);                                    // data_size code 2 = 4 bytes
  g1[1] = (int)((td0 & 0xFFFFu) << 16);                       // tensor_dim0[15:0]
  g1[2] = (int)((td0 >> 16) | ((td1 & 0xFFFFu) << 16));       // tensor_dim0[31:16] | tensor_dim1[15:0]
  g1[3] = (int)((td1 >> 16) | (tile_elems << 16));            // tensor_dim1[31:16] | tile_dim0
  g1[4] = (int)(tile_rows & 0xFFFFu);                         // tile_dim1 (tile_dim2 = 0)
  g1[5] = (int)(row_stride_elems & 0xFFFFFFFFull);            // tensor_dim0_stride[31:0]
  g1[6] = (int)((row_stride_elems >> 32) & 0xFFFFull);        // stride[47:32] (dim1_stride = 0)
  g1[7] = 0;
  i32x4 gz4 = {0, 0, 0, 0};
#if __clang_major__ >= 23
  i32x8 gz8 = {0, 0, 0, 0, 0, 0, 0, 0};
  __builtin_amdgcn_tensor_load_to_lds(g0, g1, gz4, gz4, gz8, 0);
#else
  __builtin_amdgcn_tensor_load_to_lds(g0, g1, gz4, gz4, 0);
#endif
}
#endif // USE_TDM

__global__ __launch_bounds__(TILE_X * TILE_Y)
void splatter_blender_kernel(const float* __restrict__ colors,
                             const float* __restrict__ pcs,
                             const int*   __restrict__ bmask,
                             const float* __restrict__ bgcol,
                             float*       __restrict__ out)
{
  extern __shared__ float smem[];
  float* DEP  = smem;                      // [NHALO][4] masked depths (b128 access)
  float* PRE  = smem + PRE_OFF;            // [NHALO][37] splat factors + rgb
  float* RAWC = smem + RAWC_OFF;
  float* RAWP = smem + RAWP_OFF;
  int*   RAWM = (int*)(smem + RAWM_OFF);

  const int tx = threadIdx.x, ty = threadIdx.y;
  const int tid = ty * TILE_X + tx;
  const int bx = blockIdx.x, by = blockIdx.y, n = blockIdx.z;
  const int h0 = by * TILE_Y, w0 = bx * TILE_X;

  // Full 18x10 halo window in-bounds?  (then TDM can DMA it in one shot)
  const bool interior = (USE_TDM != 0) &&
                        (bx > 0) && (by > 0) &&
                        (bx < (int)gridDim.x - 1) && (by < (int)gridDim.y - 1);

#if USE_TDM
  if (interior && ty < 2) {   // wave 0 only (threads 0..31) issues the DMA
    const long long pix0 = ((long long)(n * H_ + (h0 - 1))) * W_ + (w0 - 1);
    tdm_load_2d((unsigned)(size_t)(void*)RAWC, colors + pix0 * 12, HX * 12, HY,
                (unsigned long long)W_ * 12);
    tdm_load_2d((unsigned)(size_t)(void*)RAWP, pcs + pix0 * 12, HX * 12, HY,
                (unsigned long long)W_ * 12);
    tdm_load_2d((unsigned)(size_t)(void*)RAWM, bmask + pix0 * 4, HX * 4, HY,
                (unsigned long long)W_ * 4);
    __builtin_amdgcn_s_wait_tensorcnt(0);
  }
#endif
  __syncthreads();

  // NORM = (1+eps) / sum_d exp(-(dy^2+dx^2)/(2 sigma^2)),  sigma=0.5, eps=0.05
  const float NORMC =
      (float)(1.05 / (1.0 + 4.0 * 0.13533528323661270 + 4.0 * 0.018315638888734179));

  // ---- Stage 1: per halo point, precompute depth + separable splat factors.
  for (int idx = tid; idx < NHALO; idx += TILE_X * TILE_Y) {
    const int hh = idx / HX, ww = idx - hh * HX;
    const int gh = h0 - 1 + hh, gw = w0 - 1 + ww;
    float* pre = &PRE[idx * PSTRIDE];
    if (gh < 0 || gh >= H_ || gw < 0 || gw >= W_) {
      // zero-padded neighbor: depth 0, all weights 0 -> contributes nothing
      float4 z4; z4.x = z4.y = z4.z = z4.w = 0.0f;
      *(float4*)(DEP + idx * 4) = z4;
      #pragma unroll
      for (int j = 0; j < 36; ++j) pre[j] = 0.0f;
      continue;
    }
    float col[12], pcv[12]; int mv[4];
    if (interior) {
      const float4* rc = (const float4*)(RAWC + idx * 12);
      ((float4*)col)[0] = rc[0]; ((float4*)col)[1] = rc[1]; ((float4*)col)[2] = rc[2];
      const float4* rp = (const float4*)(RAWP + idx * 12);
      ((float4*)pcv)[0] = rp[0]; ((float4*)pcv)[1] = rp[1]; ((float4*)pcv)[2] = rp[2];
      ((int4*)mv)[0] = *(const int4*)(RAWM + idx * 4);
    } else {
      const long long pix = ((long long)(n * H_ + gh)) * W_ + gw;
      const float4* gc = (const float4*)(colors + pix * 12);
      ((float4*)col)[0] = gc[0]; ((float4*)col)[1] = gc[1]; ((float4*)col)[2] = gc[2];
      const float4* gp = (const float4*)(pcs + pix * 12);
      ((float4*)pcv)[0] = gp[0]; ((float4*)pcv)[1] = gp[1]; ((float4*)pcv)[2] = gp[2];
      ((int4*)mv)[0] = *(const int4*)(bmask + pix * 4);
    }
    float dep[4];
    #pragma unroll
    for (int k = 0; k < K_; ++k) {
      const bool m = (mv[k] != 0);
      dep[k] = m ? 1.0f : pcv[k * 3 + 2];       // masked depth
      const float alpha = m ? 0.0f : 1.0f;      // alpha in {0,1}; folded into wy
      const float fx = pcv[k * 3 + 0] - (float)gw;
      const float fy = pcv[k * 3 + 1] - (float)gh;
      float* f = pre + k * 9;
      #pragma unroll
      for (int j = 0; j < 3; ++j) {             // exponent = -d2/(2*0.25) = -2*d2
        const float ex = fx - (float)(j - 1);
        f[j]     = __expf(-2.0f * ex * ex);
        const float ey = fy - (float)(j - 1);
        f[3 + j] = NORMC * alpha * __expf(-2.0f * ey * ey);
      }
      f[6] = col[k * 3 + 0]; f[7] = col[k * 3 + 1]; f[8] = col[k * 3 + 2];
    }
    float4 d4; d4.x = dep[0]; d4.y = dep[1]; d4.z = dep[2]; d4.w = dep[3];
    *(float4*)(DEP + idx * 4) = d4;             // single ds_store_b128
  }
  __syncthreads();

  // ---- Stage 2: per pixel, occlusion ids + bucketed splat accumulation.
  const int qh = ty + 1, qw = tx + 1;
  const float4 dq4 = *(const float4*)(DEP + (qh * HX + qw) * 4);
  float dq[4] = {dq4.x, dq4.y, dq4.z, dq4.w};

  float aR[3] = {0, 0, 0}, aG[3] = {0, 0, 0}, aB[3] = {0, 0, 0}, aW[3] = {0, 0, 0};
  const int DYS[9] = {-1, -1, -1, 0, 0, 0, 1, 1, 1};
  const int DXS[9] = {-1, 0, 1, -1, 0, 1, -1, 0, 1};

  #pragma unroll
  for (int d = 0; d < 9; ++d) {
    const int dy = DYS[d], dx = DXS[d];
    const int pIdx = (qh - dy) * HX + (qw - dx);
    const float4 dp4 = *(const float4*)(DEP + pIdx * 4);   // one ds_load_b128
    const float dp[4] = {dp4.x, dp4.y, dp4.z, dp4.w};
    const float* ppre = &PRE[pIdx * PSTRIDE];

    // occ[q,d]: argmin over k (first-min tie-break, strict <, like jnp.argmin)
    const float qtop = dq[0];
    float qpd = fabsf(dp[0] - qtop); int qpid = 0;
    float pqd = fabsf(dp[0] - dq[0]); int pqid = 0;
    #pragma unroll
    for (int k = 1; k < 4; ++k) {
      const float v1 = fabsf(dp[k] - qtop);
      if (v1 < qpd) { qpd = v1; qpid = k; }
      const float v2 = fabsf(dp[0] - dq[k]);
      if (v2 < pqd) { pqd = v2; pqid = k; }
    }
    const int occ = (qpd < pqd) ? qpid : -pqid;

    const int jx = dx + 1, jy = dy + 1;
    #pragma unroll
    for (int k = 0; k < 4; ++k) {
      const float* f = ppre + k * 9;
      const float w = f[jx] * f[3 + jy];        // separable Gaussian * NORM * alpha
      const float wbg = (k > occ)  ? w : 0.0f;  // bucket 0: background
      const float wsf = (k == occ) ? w : 0.0f;  // bucket 1: surface
      const float wfg = (k < occ)  ? w : 0.0f;  // bucket 2: foreground
      const float r = f[6], g = f[7], b = f[8];
      aR[0] += wbg * r; aG[0] += wbg * g; aB[0] += wbg * b; aW[0] += wbg;
      aR[1] += wsf * r; aG[1] += wsf * g; aB[1] += wsf * b; aW[1] += wsf;
      aR[2] += wfg * r; aG[2] += wfg * g; aB[2] += wfg * b; aW[2] += wfg;
    }
  }

  // ---- Stage 3: normalize buckets and composite over background.
  // (alpha channel == weight channel since per-point alpha is 0/1)
  float ox = bgcol[0], oy = bgcol[1], oz = bgcol[2], ow = 1.0f;
  #pragma unroll
  for (int i = 0; i < 3; ++i) {                 // order: bg, surface, fg
    const float inv = 1.0f / fmaxf(aW[i], 1.0f);
    const float lr = aR[i] * inv, lg = aG[i] * inv, lb = aB[i] * inv, la = aW[i] * inv;
    const float t = 1.0f - la;
    ox = lr + t * ox; oy = lg + t * oy; oz = lb + t * oz; ow = la + t * ow;
  }

  const long long opix = ((long long)(n * H_ + (h0 + ty))) * W_ + (w0 + tx);
  float4 res; res.x = ox; res.y = oy; res.z = oz; res.w = ow;
  *(float4*)(out + opix * 4) = res;
}

extern "C" void kernel_launch(void* const* d_in, const int* in_sizes, int n_in,
                              void* d_out, int out_size, void* d_ws, size_t ws_size,
                              hipStream_t stream) {
  (void)in_sizes; (void)n_in; (void)out_size; (void)d_ws; (void)ws_size;
  const float* colors = (const float*)d_in[0];
  const float* pcs    = (const float*)d_in[1];
  const int*   bmask  = (const int*)d_in[2];
  const float* bgc    = (const float*)d_in[3];
  float* out = (float*)d_out;

  dim3 grid(W_ / TILE_X, H_ / TILE_Y, N_);   // (32, 64, 2)
  dim3 block(TILE_X, TILE_Y);                // 128 threads = 4 wave32s
  hipLaunchKernelGGL(splatter_blender_kernel, grid, block, SMEM_BYTES, stream,
                     colors, pcs, bmask, bgc, out);
}